// GaussianMixture_86964497809967
// MI455X (gfx1250) — compile-verified
//
#include <hip/hip_runtime.h>
#include <math.h>

typedef __attribute__((ext_vector_type(2))) float v2f;
typedef __attribute__((ext_vector_type(8))) float v8f;

#define NPTS 262144
#define KC 8
#define DD 64

// Workspace layout in floats:
//   P[k][ep][d][2]  : 8*32*64*2 = 32768 floats  (B-fragment-packed M^T, e-pairs)
//   t[k][d]         : 8*64      = 512 floats    (t_k = M_k mu_k)
//   cconst[k]       : 8 floats                  (log-norm const + log pi)
#define T_OFF 32768
#define C_OFF 33280
#define SMEM_FLOATS 33288

// ---------------------------------------------------------------------------
// Prep: per component k, Cholesky var = C C^T, M = C^{-1} (lower), pack M^T
// for WMMA B fragments, compute t_k = M mu_k and the log constant.
// ---------------------------------------------------------------------------
__global__ __launch_bounds__(64) void gmm_prep(const float* __restrict__ mu,
                                               const float* __restrict__ var,
                                               const float* __restrict__ pi,
                                               float* __restrict__ ws) {
    const int k = blockIdx.x;
    const int tid = threadIdx.x;          // 64 threads
    __shared__ float Cm[DD][DD];
    __shared__ float Mi[DD][DD];

    for (int i = tid; i < DD * DD; i += 64)
        Cm[i >> 6][i & 63] = var[(size_t)k * DD * DD + i];
    __syncthreads();

    // In-place lower Cholesky
    for (int j = 0; j < DD; ++j) {
        if (tid == j) {
            float s = Cm[j][j];
            for (int p = 0; p < j; ++p) s -= Cm[j][p] * Cm[j][p];
            Cm[j][j] = sqrtf(fmaxf(s, 1e-30f));
        }
        __syncthreads();
        if (tid > j) {
            float s = Cm[tid][j];
            for (int p = 0; p < j; ++p) s -= Cm[tid][p] * Cm[j][p];
            Cm[tid][j] = s / Cm[j][j];
        }
        __syncthreads();
    }

    // Invert lower-triangular C: thread 'c' solves column c of M = C^{-1}
    {
        const int c = tid;
        for (int r = 0; r < DD; ++r) {
            float v;
            if (r < c) {
                v = 0.0f;
            } else {
                v = (r == c) ? 1.0f : 0.0f;
                for (int p = c; p < r; ++p) v -= Cm[r][p] * Mi[p][c];
                v /= Cm[r][r];
            }
            Mi[r][c] = v;
        }
    }
    __syncthreads();

    // Pack B = M^T as consecutive-e pairs: P[k][ep][d] = (M[d][2ep], M[d][2ep+1])
    {
        const int d = tid;
        for (int ep = 0; ep < 32; ++ep) {
            float2 val = make_float2(Mi[d][2 * ep], Mi[d][2 * ep + 1]);
            ((float2*)ws)[((size_t)k * 32 + ep) * DD + d] = val;
        }
        // t_k[d] = sum_e M[d][e] * mu[k][e]
        float tv = 0.0f;
        for (int e = 0; e < DD; ++e) tv += Mi[d][e] * mu[k * DD + e];
        ws[T_OFF + k * DD + d] = tv;
    }
    if (tid == 0) {
        // 0.5*logdet(prec) = sum log Mi[j][j]
        float ld = 0.0f;
        for (int j = 0; j < DD; ++j) ld += logf(Mi[j][j]);
        ws[C_OFF + k] = -0.5f * (float)DD * logf(2.0f * (float)M_PI) + ld + logf(pi[k]);
    }
}

// ---------------------------------------------------------------------------
// Main: each wave handles 16 points. Z = X * M^T via v_wmma_f32_16x16x4_f32,
// quad = rowsum((Z - t)^2), then softmax over K.
// ---------------------------------------------------------------------------
__global__ __launch_bounds__(256) void gmm_post(const float* __restrict__ x,
                                                const float* __restrict__ ws,
                                                float* __restrict__ out) {
    __shared__ float smem[SMEM_FLOATS];   // 133,152 B: P + t + const
    for (int i = threadIdx.x; i < SMEM_FLOATS; i += 256) smem[i] = ws[i];
    __syncthreads();

    const int lane = threadIdx.x & 31;
    const int wave = threadIdx.x >> 5;
    const int h = lane >> 4;      // half-wave selector (K split of A/B frags)
    const int r = lane & 15;      // row (A) / column (B,C,D) within tile
    const size_t rowbase = (size_t)blockIdx.x * 128 + (size_t)wave * 16;

    // A fragments: 16x4 f32 tiles of X rows. ISA layout: v0 = K+0 (lanes 0-15)
    // / K+2 (lanes 16-31); v1 = K+1 / K+3. Per lane: pairs at col 4s + 2h.
    const float* xrow = x + (rowbase + r) * DD + 2 * h;
    v2f a[16];
#pragma unroll
    for (int s = 0; s < 16; ++s)
        a[s] = *(const v2f*)(xrow + 4 * s);

    float sk[KC][8];   // weighted log prob for rows (i + 8h), all components

#pragma unroll
    for (int k = 0; k < KC; ++k) {
        const float* Pk = smem + k * 4096;
        v8f q = {};
#pragma unroll
        for (int nt = 0; nt < 4; ++nt) {
            v8f c = {};
#pragma unroll
            for (int s = 0; s < 16; ++s) {
                const int ep = 2 * s + h;   // e-pair index for this lane half
                v2f b = *(const v2f*)(Pk + (size_t)(ep * DD + nt * 16 + r) * 2);
                c = __builtin_amdgcn_wmma_f32_16x16x4_f32(
                        false, a[s], false, b, (short)0, c, false, false);
            }
            const float tv = smem[T_OFF + k * DD + nt * 16 + r];
#pragma unroll
            for (int i = 0; i < 8; ++i) {
                float e = c[i] - tv;
                q[i] += e * e;
            }
        }
        // Reduce across the 16 lanes of each half (sum over output dims)
#pragma unroll
        for (int i = 0; i < 8; ++i) {
            float v = q[i];
            v += __shfl_xor(v, 1, 32);
            v += __shfl_xor(v, 2, 32);
            v += __shfl_xor(v, 4, 32);
            v += __shfl_xor(v, 8, 32);
            sk[k][i] = smem[C_OFF + k] - 0.5f * v;
        }
    }

    // Softmax over K per row; lanes r<8 of each half store component r.
#pragma unroll
    for (int i = 0; i < 8; ++i) {
        float m = sk[0][i];
#pragma unroll
        for (int k = 1; k < KC; ++k) m = fmaxf(m, sk[k][i]);
        float sum = 0.0f, sv = 0.0f;
#pragma unroll
        for (int k = 0; k < KC; ++k) {
            float e = __expf(sk[k][i] - m);
            sum += e;
            sv = (k == r) ? e : sv;
        }
        if (r < KC)
            out[(rowbase + (size_t)i + 8 * h) * KC + r] = sv / sum;
    }
}

extern "C" void kernel_launch(void* const* d_in, const int* in_sizes, int n_in,
                              void* d_out, int out_size, void* d_ws, size_t ws_size,
                              hipStream_t stream) {
    (void)in_sizes; (void)n_in; (void)out_size; (void)ws_size;
    const float* x   = (const float*)d_in[0];   // (N, 64)
    const float* mu  = (const float*)d_in[1];   // (1, 8, 64)
    const float* var = (const float*)d_in[2];   // (1, 8, 64, 64)
    const float* pi  = (const float*)d_in[3];   // (1, 8, 1)
    float* ws  = (float*)d_ws;
    float* out = (float*)d_out;                 // (N, 8)

    gmm_prep<<<KC, 64, 0, stream>>>(mu, var, pi, ws);
    gmm_post<<<NPTS / 128, 256, 0, stream>>>(x, ws, out);
}